// BayesianLinear_19112604467219
// MI455X (gfx1250) — compile-verified
//
#include <hip/hip_runtime.h>
#include <math.h>

// ---------------------------------------------------------------------------
// BayesianLinear: out = X @ (Wmu + softplus(Wrho)*Weps)^T + (bmu + softplus(brho)*beps)
// X: [4096,4096] f32, W*: [4096,4096] f32, out: [4096,4096] f32
//
// Pass 1: materialize sampled W into d_ws (64 MB) once (reads 192 MB of
//         parameters exactly once -> minimal HBM traffic).
// Pass 2: FP32 WMMA GEMM (v_wmma_f32_16x16x4_f32), 128x128 tile/workgroup,
//         K-tile 32, DOUBLE-BUFFERED LDS filled by CDNA5 async copies
//         (global_load_async_to_lds_b128, ASYNCcnt), no VGPR staging.
// ---------------------------------------------------------------------------

typedef __attribute__((ext_vector_type(2))) float v2f;
typedef __attribute__((ext_vector_type(8))) float v8f;

#define N_FEAT 4096
#define TILE   128
#define KT     32
#define KS     36   // padded LDS row stride (floats): 144B rows -> 16B aligned,
                    // 36*r mod 64 distinct for r=0..15 -> conflict-free frags

__device__ __forceinline__ float softplus_eps(float r) {
    float sp = (r > 20.0f) ? r : log1pf(expf(r));
    return sp + 1e-8f;
}

// Async global -> LDS copy of 16 bytes per lane (CDNA5 TDM-adjacent path).
// ldsoff: per-lane LDS byte offset (low 32 bits of a generic LDS pointer);
// gaddr:  per-lane 64-bit global byte address.
__device__ __forceinline__ void async_copy_b128(unsigned ldsoff,
                                                unsigned long long gaddr) {
    asm volatile("global_load_async_to_lds_b128 %0, %1, off"
                 :: "v"(ldsoff), "v"(gaddr)
                 : "memory");
}

// ---------------- Kernel 1: sample weights into workspace -------------------
__global__ __launch_bounds__(256)
void sample_weights(const float4* __restrict__ mu,
                    const float4* __restrict__ rho,
                    const float4* __restrict__ eps,
                    float4* __restrict__ w, int n4) {
    int i = blockIdx.x * blockDim.x + threadIdx.x;
    if (i >= n4) return;
    float4 m = mu[i];
    float4 r = rho[i];
    float4 e = eps[i];
    float4 o;
    o.x = fmaf(softplus_eps(r.x), e.x, m.x);
    o.y = fmaf(softplus_eps(r.y), e.y, m.y);
    o.z = fmaf(softplus_eps(r.z), e.z, m.z);
    o.w = fmaf(softplus_eps(r.w), e.w, m.w);
    w[i] = o;
}

// ---------------- Kernel 2: FP32 WMMA GEMM + bias epilogue ------------------
// Block: 256 threads = 8 waves, computes a 128(M) x 128(N) tile of out.
// Wave (wm in 0..1, wn in 0..3) computes 64(M) x 32(N): 4x2 tiles of 16x16.
__global__ __launch_bounds__(256)
void gemm_wmma_f32(const float* __restrict__ X,   // [4096,4096] row-major (M,K)
                   const float* __restrict__ W,   // [4096,4096] row-major (N,K)
                   const float* __restrict__ bmu,
                   const float* __restrict__ brho,
                   const float* __restrict__ beps,
                   float* __restrict__ out) {     // [4096,4096] row-major (M,N)
    __shared__ float ldsX[2][TILE * KS];
    __shared__ float ldsW[2][TILE * KS];

    const int tid  = threadIdx.x;
    const int lane = tid & 31;
    const int wave = tid >> 5;
    const int wm   = wave >> 2;       // 0..1  -> M offset wm*64
    const int wn   = wave & 3;        // 0..3  -> N offset wn*32
    const int lo   = lane & 15;       // row within 16x16 fragment
    const int hi   = lane >> 4;       // half-wave select

    const int tileM = blockIdx.y * TILE;
    const int tileN = blockIdx.x * TILE;

    v8f c[4][2];
#pragma unroll
    for (int mi = 0; mi < 4; ++mi)
#pragma unroll
        for (int ni = 0; ni < 2; ++ni)
            c[mi][ni] = (v8f){0.f, 0.f, 0.f, 0.f, 0.f, 0.f, 0.f, 0.f};

    // Loader mapping: 2 threads per row, each covers 16 floats (4x b128)
    const int lr = tid >> 1;          // 0..127 tile row
    const int lh = (tid & 1) * 16;    // 0 or 16 column offset within K-tile

    const unsigned long long gxBase =
        (unsigned long long)(const void*)(X + (size_t)(tileM + lr) * N_FEAT + lh);
    const unsigned long long gwBase =
        (unsigned long long)(const void*)(W + (size_t)(tileN + lr) * N_FEAT + lh);
    const unsigned lxBase0 = (unsigned)(size_t)(const void*)(&ldsX[0][lr * KS + lh]);
    const unsigned lwBase0 = (unsigned)(size_t)(const void*)(&ldsW[0][lr * KS + lh]);
    const unsigned lxBase1 = (unsigned)(size_t)(const void*)(&ldsX[1][lr * KS + lh]);
    const unsigned lwBase1 = (unsigned)(size_t)(const void*)(&ldsW[1][lr * KS + lh]);

    // Issue one K-stage (8 async b128 per thread: 4 X rows-halves + 4 W)
    auto issue_stage = [&](int s, int k0) {
        const unsigned lx = s ? lxBase1 : lxBase0;
        const unsigned lw = s ? lwBase1 : lwBase0;
        const unsigned long long gx = gxBase + (unsigned)(k0 * 4);
        const unsigned long long gw = gwBase + (unsigned)(k0 * 4);
#pragma unroll
        for (int q = 0; q < 4; ++q) {
            async_copy_b128(lx + 16u * q, gx + 16ull * q);
            async_copy_b128(lw + 16u * q, gw + 16ull * q);
        }
    };

    const int NSTAGES = N_FEAT / KT;   // 128
    issue_stage(0, 0);

    for (int i = 0; i < NSTAGES; ++i) {
        const int s = i & 1;
        if (i + 1 < NSTAGES) {
            issue_stage(s ^ 1, (i + 1) * KT);
            // 8 outstanding from next stage; <=8 means current stage landed
            asm volatile("s_wait_asynccnt 0x8" ::: "memory");
        } else {
            asm volatile("s_wait_asynccnt 0x0" ::: "memory");
        }
        __syncthreads();   // publish current stage's LDS to all waves

        // Compute: 8 k-steps of 4, 8 WMMA tiles each -> 64 v_wmma per stage
        const float* ax = &ldsX[s][(wm * 64 + lo) * KS + 2 * hi];
        const float* bx = &ldsW[s][(wn * 32 + lo) * KS + 2 * hi];
#pragma unroll
        for (int kk = 0; kk < KT; kk += 4) {
            v2f a[4], b[2];
#pragma unroll
            for (int mi = 0; mi < 4; ++mi)
                a[mi] = *(const v2f*)(ax + mi * 16 * KS + kk);
#pragma unroll
            for (int ni = 0; ni < 2; ++ni)
                b[ni] = *(const v2f*)(bx + ni * 16 * KS + kk);
#pragma unroll
            for (int mi = 0; mi < 4; ++mi)
#pragma unroll
                for (int ni = 0; ni < 2; ++ni)
                    c[mi][ni] = __builtin_amdgcn_wmma_f32_16x16x4_f32(
                        /*neg_a=*/false, a[mi],
                        /*neg_b=*/false, b[ni],
                        /*c_mod=*/(short)0, c[mi][ni],
                        /*reuse_a=*/false, /*reuse_b=*/false);
        }

        __syncthreads();   // all waves done reading buffer s before it is
                           // overwritten by the issue at iteration i+1
    }

    // Epilogue: bias (sampled on the fly) + store.
    // C layout: VGPR r holds M = r + 8*hi, N = lo (per 16x16 tile).
#pragma unroll
    for (int ni = 0; ni < 2; ++ni) {
        const int col = tileN + wn * 32 + ni * 16 + lo;
        const float bv = fmaf(softplus_eps(brho[col]), beps[col], bmu[col]);
#pragma unroll
        for (int mi = 0; mi < 4; ++mi) {
            const int rbase = tileM + wm * 64 + mi * 16 + hi * 8;
#pragma unroll
            for (int r = 0; r < 8; ++r)
                out[(size_t)(rbase + r) * N_FEAT + col] = c[mi][ni][r] + bv;
        }
    }
}

// ---------------------------------------------------------------------------
extern "C" void kernel_launch(void* const* d_in, const int* in_sizes, int n_in,
                              void* d_out, int out_size, void* d_ws, size_t ws_size,
                              hipStream_t stream) {
    const float* x    = (const float*)d_in[0];
    const float* wmu  = (const float*)d_in[1];
    const float* wrho = (const float*)d_in[2];
    const float* bmu  = (const float*)d_in[3];
    const float* brho = (const float*)d_in[4];
    const float* weps = (const float*)d_in[5];
    const float* beps = (const float*)d_in[6];
    float* out = (float*)d_out;

    // Workspace: sampled weight matrix, 4096*4096*4 = 64 MB
    float* W = (float*)d_ws;

    const int n4 = (N_FEAT * N_FEAT) / 4;
    sample_weights<<<n4 / 256, 256, 0, stream>>>(
        (const float4*)wmu, (const float4*)wrho, (const float4*)weps,
        (float4*)W, n4);

    dim3 grid(N_FEAT / TILE, N_FEAT / TILE);  // 32 x 32 workgroups
    gemm_wmma_f32<<<grid, 256, 0, stream>>>(x, W, bmu, brho, beps, out);
}